// G2LFormer_15126874817104
// MI455X (gfx1250) — compile-verified
//
#include <hip/hip_runtime.h>
#include <cstdint>
#include <cstddef>

#define DFEAT 128
#define BN_EPS_F 1e-5f
#define AGG_EPS_F 1e-6f

typedef float v2f __attribute__((ext_vector_type(2)));
typedef float v8f __attribute__((ext_vector_type(8)));

#define ATOMIC_ADD_G(p, v) (void)__hip_atomic_fetch_add((p), (v), __ATOMIC_RELAXED, __HIP_MEMORY_SCOPE_AGENT)
#define ATOMIC_ADD_S(p, v) (void)__hip_atomic_fetch_add((p), (v), __ATOMIC_RELAXED, __HIP_MEMORY_SCOPE_WORKGROUP)

__device__ __forceinline__ float relu_(float v) { return v > 0.f ? v : 0.f; }

// ---------------------------------------------------------------------------
// Generic fp32 WMMA GEMM:  C[M,N] = A[M,K] @ W[K,N] + bias   (optional ReLU)
// Block = 64 threads (2 wave32), 32 output rows per block, n-tiles of 16.
// Uses v_wmma_f32_16x16x4_f32; A-tile and W^T n-tile staged in LDS.
// ---------------------------------------------------------------------------
template <int K, int NTILES, bool RELU>
__global__ __launch_bounds__(64) void wmma_gemm_kernel(
    const float* __restrict__ A, const float* __restrict__ W,
    const float* __restrict__ bias, float* __restrict__ C, int M)
{
    constexpr int N  = NTILES * 16;
    constexpr int KP = K + 2;                 // pad to dodge LDS bank conflicts
    __shared__ float As[32 * KP];
    __shared__ float Bs[16 * KP];

    const int tid  = threadIdx.x;
    const int lane = tid & 31;
    const int wave = tid >> 5;
    const int l16  = lane & 15;
    const int kk   = (lane < 16) ? 0 : 2;     // A/B k-half per ISA frag layout
    const int mofs = (lane < 16) ? 0 : 8;     // C/D row-half per ISA frag layout
    const int rowB = blockIdx.x * 32;

    // Stage 32 x K A-tile (row-clamped for tail blocks; results are masked on store)
    for (int i = tid; i < 32 * (K / 4); i += 64) {
        int r = i / (K / 4), c4 = i % (K / 4);
        int gr = rowB + r; if (gr >= M) gr = M - 1;
        float4 v = reinterpret_cast<const float4*>(A + (size_t)gr * K)[c4];
        float* d = &As[r * KP + c4 * 4];
        d[0] = v.x; d[1] = v.y; d[2] = v.z; d[3] = v.w;
    }

    const float* aRow = &As[(wave * 16 + l16) * KP];

    for (int t = 0; t < NTILES; ++t) {
        const int n0 = t * 16;
        __syncthreads();
        // Stage W^T tile: Bs[n][k] = W[k][n0+n]  (coalesced global reads)
        for (int k = tid >> 4; k < K; k += 4)
            Bs[(tid & 15) * KP + k] = W[(size_t)k * N + n0 + (tid & 15)];
        __syncthreads();

        v8f acc = {0.f, 0.f, 0.f, 0.f, 0.f, 0.f, 0.f, 0.f};
        const float* bRow = &Bs[l16 * KP];
#pragma unroll
        for (int k0 = 0; k0 < K; k0 += 4) {
            v2f a, b;
            a.x = aRow[k0 + kk]; a.y = aRow[k0 + kk + 1];
            b.x = bRow[k0 + kk]; b.y = bRow[k0 + kk + 1];
            acc = __builtin_amdgcn_wmma_f32_16x16x4_f32(
                false, a, false, b, (short)0, acc, false, false);
        }

        const float bv = bias[n0 + l16];
#pragma unroll
        for (int v = 0; v < 8; ++v) {
            int gr = rowB + wave * 16 + v + mofs;
            if (gr < M) {
                float val = acc[v] + bv;
                if (RELU) val = relu_(val);
                C[(size_t)gr * N + n0 + l16] = val;
            }
        }
    }
}

// ---------------------------------------------------------------------------
// Fused edge kernel:  Ce = e @ WC + bC  (WMMA)  then per element
//   e_ij = Ce + Dx[dst] + Ex[src]   (stored into d_out e-region)
//   sigma = sigmoid(e_ij)
//   num[dst] += sigma * Bx[src];  den[dst] += sigma      (L2-resident atomics)
//   per-column sum / sumsq of e_ij for edge BatchNorm (LDS -> global atomics)
// ---------------------------------------------------------------------------
__global__ __launch_bounds__(64) void edge_fused_kernel(
    const float* __restrict__ Ee, const float* __restrict__ W,
    const float* __restrict__ bias, const int* __restrict__ ei,
    const float* __restrict__ Bx, const float* __restrict__ Dx,
    const float* __restrict__ Ex, float* __restrict__ num,
    float* __restrict__ den, float* __restrict__ eij,
    float* __restrict__ eSum, float* __restrict__ eSq, int M)
{
    constexpr int K = 128, KP = 130;
    __shared__ float As[32 * KP];
    __shared__ float Bs[16 * KP];
    __shared__ float cSum[128];
    __shared__ float cSq[128];

    const int tid  = threadIdx.x;
    const int lane = tid & 31;
    const int wave = tid >> 5;
    const int l16  = lane & 15;
    const int kk   = (lane < 16) ? 0 : 2;
    const int mofs = (lane < 16) ? 0 : 8;
    const int rowB = blockIdx.x * 32;

    for (int i = tid; i < 128; i += 64) { cSum[i] = 0.f; cSq[i] = 0.f; }

    for (int i = tid; i < 32 * (K / 4); i += 64) {
        int r = i / (K / 4), c4 = i % (K / 4);
        int gr = rowB + r; if (gr >= M) gr = M - 1;
        float4 v = reinterpret_cast<const float4*>(Ee + (size_t)gr * K)[c4];
        float* d = &As[r * KP + c4 * 4];
        d[0] = v.x; d[1] = v.y; d[2] = v.z; d[3] = v.w;
    }

    // Preload this lane's 8 edge endpoints (edge_index: src = ei[0:E], dst = ei[E:2E])
    int sE[8], dE[8];
#pragma unroll
    for (int v = 0; v < 8; ++v) {
        int edge = rowB + wave * 16 + v + mofs;
        int ec = (edge < M) ? edge : (M - 1);
        sE[v] = ei[ec];
        dE[v] = ei[M + ec];
    }

    const float* aRow = &As[(wave * 16 + l16) * KP];

    for (int t = 0; t < 8; ++t) {
        const int n0 = t * 16;
        __syncthreads();
        for (int k = tid >> 4; k < K; k += 4)
            Bs[(tid & 15) * KP + k] = W[(size_t)k * DFEAT + n0 + (tid & 15)];
        __syncthreads();

        v8f acc = {0.f, 0.f, 0.f, 0.f, 0.f, 0.f, 0.f, 0.f};
        const float* bRow = &Bs[l16 * KP];
#pragma unroll
        for (int k0 = 0; k0 < K; k0 += 4) {
            v2f a, b;
            a.x = aRow[k0 + kk]; a.y = aRow[k0 + kk + 1];
            b.x = bRow[k0 + kk]; b.y = bRow[k0 + kk + 1];
            acc = __builtin_amdgcn_wmma_f32_16x16x4_f32(
                false, a, false, b, (short)0, acc, false, false);
        }

        const int n = n0 + l16;
        const float bv = bias[n];
        float ls = 0.f, lq = 0.f;
#pragma unroll
        for (int v = 0; v < 8; ++v) {
            int edge = rowB + wave * 16 + v + mofs;
            if (edge < M) {
                float val = acc[v] + bv
                          + Dx[(size_t)dE[v] * DFEAT + n]
                          + Ex[(size_t)sE[v] * DFEAT + n];
                eij[(size_t)edge * DFEAT + n] = val;
                float sg = 1.f / (1.f + __expf(-val));
                ATOMIC_ADD_G(&num[(size_t)dE[v] * DFEAT + n],
                             sg * Bx[(size_t)sE[v] * DFEAT + n]);
                ATOMIC_ADD_G(&den[(size_t)dE[v] * DFEAT + n], sg);
                ls += val; lq += val * val;
            }
        }
        ATOMIC_ADD_S(&cSum[n], ls);
        ATOMIC_ADD_S(&cSq[n],  lq);
    }
    __syncthreads();
    for (int c = tid; c < 128; c += 64) {
        ATOMIC_ADD_G(&eSum[c], cSum[c]);
        ATOMIC_ADD_G(&eSq[c],  cSq[c]);
    }
}

// ---------------------------------------------------------------------------
// Elementwise / reduction helper kernels
// ---------------------------------------------------------------------------
__global__ void zero_kernel(float* __restrict__ p, size_t n)
{
    size_t i = (size_t)blockIdx.x * blockDim.x + threadIdx.x;
    size_t s = (size_t)gridDim.x * blockDim.x;
    for (; i < n; i += s) p[i] = 0.f;
}

// t := t + num/(den+eps); accumulate per-column sum/sumsq
__global__ void node_aggr_kernel(float* t, const float* __restrict__ num,
                                 const float* __restrict__ den,
                                 float* __restrict__ s1, float* __restrict__ q1,
                                 size_t n)
{
    __shared__ float cs[128], cq[128];
    for (int i = threadIdx.x; i < 128; i += blockDim.x) { cs[i] = 0.f; cq[i] = 0.f; }
    __syncthreads();
    size_t i = (size_t)blockIdx.x * blockDim.x + threadIdx.x;
    size_t s = (size_t)gridDim.x * blockDim.x;
    for (; i < n; i += s) {
        float v = t[i] + num[i] / (den[i] + AGG_EPS_F);
        t[i] = v;
        int c = (int)(i & 127);
        ATOMIC_ADD_S(&cs[c], v);
        ATOMIC_ADD_S(&cq[c], v * v);
    }
    __syncthreads();
    for (int c = threadIdx.x; c < 128; c += blockDim.x) {
        ATOMIC_ADD_G(&s1[c], cs[c]);
        ATOMIC_ADD_G(&q1[c], cq[c]);
    }
}

// scale = gamma * rsqrt(var + eps);  shift = beta - mean*scale   (1 block / 128 thr)
__global__ void bn_finalize_kernel(const float* __restrict__ sum,
                                   const float* __restrict__ sq,
                                   const float* __restrict__ gamma,
                                   const float* __restrict__ beta,
                                   float invCount, float* __restrict__ scale,
                                   float* __restrict__ shift)
{
    int c = threadIdx.x;
    if (c < 128) {
        float m   = sum[c] * invCount;
        float var = sq[c] * invCount - m * m;
        float sc  = gamma[c] * rsqrtf(var + BN_EPS_F);
        scale[c] = sc;
        shift[c] = beta[c] - m * sc;
    }
}

// e_out (in place over e_ij in d_out): out = e + relu(eij*scale + shift), float4
__global__ void edge_out_kernel(float* eij, const float* __restrict__ e,
                                const float* __restrict__ scale,
                                const float* __restrict__ shift, size_t n4)
{
    size_t i = (size_t)blockIdx.x * blockDim.x + threadIdx.x;
    size_t s = (size_t)gridDim.x * blockDim.x;
    for (; i < n4; i += s) {
        float4 v  = reinterpret_cast<float4*>(eij)[i];
        float4 ev = reinterpret_cast<const float4*>(e)[i];
        int c = (int)(i & 31) * 4;  // 32 float4 per 128-wide row
        v.x = ev.x + relu_(v.x * scale[c + 0] + shift[c + 0]);
        v.y = ev.y + relu_(v.y * scale[c + 1] + shift[c + 1]);
        v.z = ev.z + relu_(v.z * scale[c + 2] + shift[c + 2]);
        v.w = ev.w + relu_(v.w * scale[c + 3] + shift[c + 3]);
        reinterpret_cast<float4*>(eij)[i] = v;
    }
}

// b := x + relu(b*scale + shift); accumulate stats of result
__global__ void node_x1_kernel(float* b, const float* __restrict__ x,
                               const float* __restrict__ scale,
                               const float* __restrict__ shift,
                               float* __restrict__ s1, float* __restrict__ q1,
                               size_t n)
{
    __shared__ float cs[128], cq[128];
    for (int i = threadIdx.x; i < 128; i += blockDim.x) { cs[i] = 0.f; cq[i] = 0.f; }
    __syncthreads();
    size_t i = (size_t)blockIdx.x * blockDim.x + threadIdx.x;
    size_t s = (size_t)gridDim.x * blockDim.x;
    for (; i < n; i += s) {
        int c = (int)(i & 127);
        float v = x[i] + relu_(b[i] * scale[c] + shift[c]);
        b[i] = v;
        ATOMIC_ADD_S(&cs[c], v);
        ATOMIC_ADD_S(&cq[c], v * v);
    }
    __syncthreads();
    for (int c = threadIdx.x; c < 128; c += blockDim.x) {
        ATOMIC_ADD_G(&s1[c], cs[c]);
        ATOMIC_ADD_G(&q1[c], cq[c]);
    }
}

// b := b + zr; accumulate stats of result
__global__ void node_z_kernel(float* b, const float* __restrict__ zr,
                              float* __restrict__ s1, float* __restrict__ q1,
                              size_t n)
{
    __shared__ float cs[128], cq[128];
    for (int i = threadIdx.x; i < 128; i += blockDim.x) { cs[i] = 0.f; cq[i] = 0.f; }
    __syncthreads();
    size_t i = (size_t)blockIdx.x * blockDim.x + threadIdx.x;
    size_t s = (size_t)gridDim.x * blockDim.x;
    for (; i < n; i += s) {
        float v = b[i] + zr[i];
        b[i] = v;
        int c = (int)(i & 127);
        ATOMIC_ADD_S(&cs[c], v);
        ATOMIC_ADD_S(&cq[c], v * v);
    }
    __syncthreads();
    for (int c = threadIdx.x; c < 128; c += blockDim.x) {
        ATOMIC_ADD_G(&s1[c], cs[c]);
        ATOMIC_ADD_G(&q1[c], cq[c]);
    }
}

// out := in*scale + shift   (in == out allowed)
__global__ void scale_kernel(const float* in, const float* __restrict__ scale,
                             const float* __restrict__ shift, float* out, size_t n)
{
    size_t i = (size_t)blockIdx.x * blockDim.x + threadIdx.x;
    size_t s = (size_t)gridDim.x * blockDim.x;
    for (; i < n; i += s) {
        int c = (int)(i & 127);
        out[i] = in[i] * scale[c] + shift[c];
    }
}

// ---------------------------------------------------------------------------
extern "C" void kernel_launch(void* const* d_in, const int* in_sizes, int n_in,
                              void* d_out, int out_size, void* d_ws, size_t ws_size,
                              hipStream_t stream)
{
    (void)n_in; (void)out_size; (void)ws_size;

    const float* x    = (const float*)d_in[0];
    const float* e    = (const float*)d_in[1];
    const int*   ei   = (const int*)d_in[2];
    const float* WA   = (const float*)d_in[3];  const float* bA   = (const float*)d_in[4];
    const float* WB   = (const float*)d_in[5];  const float* bB   = (const float*)d_in[6];
    const float* WC   = (const float*)d_in[7];  const float* bC   = (const float*)d_in[8];
    const float* WD   = (const float*)d_in[9];  const float* bD   = (const float*)d_in[10];
    const float* WE   = (const float*)d_in[11]; const float* bE   = (const float*)d_in[12];
    const float* g_x  = (const float*)d_in[13]; const float* be_x = (const float*)d_in[14];
    const float* g_e  = (const float*)d_in[15]; const float* be_e = (const float*)d_in[16];
    const float* Wf1  = (const float*)d_in[17]; const float* bf1  = (const float*)d_in[18];
    const float* Wf2  = (const float*)d_in[19]; const float* bf2  = (const float*)d_in[20];
    const float* g_n1 = (const float*)d_in[21]; const float* be_n1 = (const float*)d_in[22];
    const float* g_n2 = (const float*)d_in[23]; const float* be_n2 = (const float*)d_in[24];

    const int NN = in_sizes[0] / DFEAT;   // 50000 nodes
    const int NE = in_sizes[1] / DFEAT;   // 600000 edges
    const size_t NM = (size_t)NN * DFEAT; // node matrix elements

    // Workspace layout (floats): 6*NM + 2048  (~154 MB)
    float* f    = (float*)d_ws;
    float* buf0 = f;            // Ax -> t -> x1 -> x2 -> z  (in-place chain)
    float* bufB = f + 1 * NM;   // Bx, later zraw
    float* bufD = f + 2 * NM;   // Dx, later h (spans bufD+bufE: NN x 256)
    float* bufE = f + 3 * NM;   // Ex
    float* num  = f + 4 * NM;
    float* den  = f + 5 * NM;
    float* st   = f + 6 * NM;   // 4 stat sets x [sum|sq|scale|shift] x 128
    float* eS  = st;
    float* xS  = st + 512;
    float* n1S = st + 1024;
    float* n2S = st + 1536;
    float* h    = bufD;         // [NN, 256]
    float* zraw = bufB;
    float* xout = (float*)d_out;
    float* eij  = (float*)d_out + NM;  // e_ij staged in e_out slot, finalized in place

    // 1) zero atomics accumulators (num, den, stats are contiguous)
    zero_kernel<<<2048, 256, 0, stream>>>(num, 2 * NM + 2048);

    // 2) node GEMMs: Ax, Bx, Dx, Ex
    const int gN = (NN + 31) / 32;
    wmma_gemm_kernel<128, 8, false><<<gN, 64, 0, stream>>>(x, WA, bA, buf0, NN);
    wmma_gemm_kernel<128, 8, false><<<gN, 64, 0, stream>>>(x, WB, bB, bufB, NN);
    wmma_gemm_kernel<128, 8, false><<<gN, 64, 0, stream>>>(x, WD, bD, bufD, NN);
    wmma_gemm_kernel<128, 8, false><<<gN, 64, 0, stream>>>(x, WE, bE, bufE, NN);

    // 3) fused edge GEMM + message + sigmoid + scatter + edge-BN stats
    const int gE = (NE + 31) / 32;
    edge_fused_kernel<<<gE, 64, 0, stream>>>(e, WC, bC, ei, bufB, bufD, bufE,
                                             num, den, eij, eS, eS + 128, NE);

    // 4) t = Ax + num/(den+eps)  (+ node-BN stats); BN scale/shift prep
    node_aggr_kernel<<<1024, 256, 0, stream>>>(buf0, num, den, xS, xS + 128, NM);
    bn_finalize_kernel<<<1, 128, 0, stream>>>(eS, eS + 128, g_e, be_e,
                                              1.f / (float)NE, eS + 256, eS + 384);
    bn_finalize_kernel<<<1, 128, 0, stream>>>(xS, xS + 128, g_x, be_x,
                                              1.f / (float)NN, xS + 256, xS + 384);

    // 5) e_out = e + relu(bn(e_ij))  (in place in d_out)
    edge_out_kernel<<<4096, 256, 0, stream>>>(eij, e, eS + 256, eS + 384,
                                              (size_t)NE * (DFEAT / 4));

    // 6) x1 = x + relu(bn_x(t));  bn_n1 stats -> x2 = bn_n1(x1)
    node_x1_kernel<<<1024, 256, 0, stream>>>(buf0, x, xS + 256, xS + 384,
                                             n1S, n1S + 128, NM);
    bn_finalize_kernel<<<1, 128, 0, stream>>>(n1S, n1S + 128, g_n1, be_n1,
                                              1.f / (float)NN, n1S + 256, n1S + 384);
    scale_kernel<<<1024, 256, 0, stream>>>(buf0, n1S + 256, n1S + 384, buf0, NM);

    // 7) FFN: h = relu(x2@Wf1 + bf1);  zraw = h@Wf2 + bf2
    wmma_gemm_kernel<128, 16, true ><<<gN, 64, 0, stream>>>(buf0, Wf1, bf1, h, NN);
    wmma_gemm_kernel<256, 8, false><<<gN, 64, 0, stream>>>(h, Wf2, bf2, zraw, NN);

    // 8) z = x2 + zraw (+ bn_n2 stats); out_x = bn_n2(z)
    node_z_kernel<<<1024, 256, 0, stream>>>(buf0, zraw, n2S, n2S + 128, NM);
    bn_finalize_kernel<<<1, 128, 0, stream>>>(n2S, n2S + 128, g_n2, be_n2,
                                              1.f / (float)NN, n2S + 256, n2S + 384);
    scale_kernel<<<1024, 256, 0, stream>>>(buf0, n2S + 256, n2S + 384, xout, NM);
}